// GAT_90718299226293
// MI455X (gfx1250) — compile-verified
//
#include <hip/hip_runtime.h>
#include <math.h>

#define NEG_SLOPE 0.2f

typedef __attribute__((ext_vector_type(2))) float v2f;
typedef __attribute__((ext_vector_type(8))) float v8f;

// ---------------------------------------------------------------- utilities
__global__ __launch_bounds__(256) void fill_f32(float* p, float v, long n) {
    long i = (long)blockIdx.x * blockDim.x + threadIdx.x;
    if (i < n) p[i] = v;
}

__global__ __launch_bounds__(256) void init_out_bias(float* out, const float* __restrict__ bias,
                                                     long total, int c) {
    long i = (long)blockIdx.x * blockDim.x + threadIdx.x;
    if (i < total) out[i] = bias[i % c];
}

// ------------------------------------------------- WMMA GEMM: Y = X*W + bias
// X: [nrowtiles*16, 128], W: [128, KOUT], bias: [KOUT], Y: [nrowtiles*16, KOUT]
// One wave computes one 16x16 tile with V_WMMA_F32_16X16X4_F32 (full fp32).
template<int KOUT>
__global__ __launch_bounds__(256)
void gemm_xw_bias(const float* __restrict__ X, const float* __restrict__ W,
                  const float* __restrict__ bias, float* __restrict__ Y,
                  int nrowtiles) {
    constexpr int K = 128;
    __shared__ float wlds[K * KOUT];
    for (int i = threadIdx.x; i < K * KOUT; i += 256) wlds[i] = W[i];
    __syncthreads();

    constexpr int COL_TILES    = KOUT / 16;      // 8 (layer1) or 4 (layer2)
    constexpr int RT_PER_BLOCK = 8 / COL_TILES;  // 8 waves per block
    const int wave = threadIdx.x >> 5;
    const int lane = threadIdx.x & 31;
    const int rt   = blockIdx.x * RT_PER_BLOCK + wave / COL_TILES;
    if (rt >= nrowtiles) return;
    const int ct  = wave % COL_TILES;
    const int m   = lane & 15;            // row (A) / col (B,C,D) within tile
    const int kb  = (lane >> 4) << 1;     // lanes 16-31 hold K+2,K+3
    const int col = ct * 16 + m;
    const float* xrow = X + (size_t)(rt * 16 + m) * K;

    v8f acc = {0.f, 0.f, 0.f, 0.f, 0.f, 0.f, 0.f, 0.f};
    #pragma unroll 8
    for (int k = 0; k < K; k += 4) {
        // A frag (16x4 f32): lane m, v0=A[m][k+kb], v1=A[m][k+kb+1]
        v2f a = *(const v2f*)(xrow + k + kb);
        // B frag (4x16 f32): lane n=col, v0=B[k+kb][n], v1=B[k+kb+1][n]
        v2f b;
        b.x = wlds[(k + kb)     * KOUT + col];
        b.y = wlds[(k + kb + 1) * KOUT + col];
        acc = __builtin_amdgcn_wmma_f32_16x16x4_f32(false, a, false, b,
                                                    (short)0, acc, false, false);
    }
    // C/D layout: VGPR v -> row v (+8 for lanes 16-31), col = lane&15
    const float bv = bias[col];
    float* yp = Y + (size_t)(rt * 16 + ((lane >> 4) << 3)) * KOUT + col;
    #pragma unroll
    for (int v = 0; v < 8; ++v) yp[(size_t)v * KOUT] = acc[v] + bv;
}

// ------------------------------------------------------- layer-1 edge passes
// wave per edge; lane handles 4 channels (channels [h*16+c], 4 lanes per head)
__global__ __launch_bounds__(256)
void edge_alpha1(const float* __restrict__ xl, const float* __restrict__ xr,
                 const int* __restrict__ ei, const float* __restrict__ att,
                 float* __restrict__ alpha, float* __restrict__ amax,
                 int E, int N) {
    const int e = blockIdx.x * 8 + (threadIdx.x >> 5);
    const int lane = threadIdx.x & 31;
    if (e >= E + N) return;
    int s, d;
    if (e < E) { s = ei[e]; d = ei[E + e]; } else { s = d = e - E; }
    const int c0 = lane * 4;
    const float4 a = *(const float4*)(xl + (size_t)s * 128 + c0);
    const float4 b = *(const float4*)(xr + (size_t)d * 128 + c0);
    const float4 t = *(const float4*)(att + c0);
    float m, sum = 0.f;
    m = a.x + b.x; m = m > 0.f ? m : NEG_SLOPE * m; sum += m * t.x;
    m = a.y + b.y; m = m > 0.f ? m : NEG_SLOPE * m; sum += m * t.y;
    m = a.z + b.z; m = m > 0.f ? m : NEG_SLOPE * m; sum += m * t.z;
    m = a.w + b.w; m = m > 0.f ? m : NEG_SLOPE * m; sum += m * t.w;
    sum += __shfl_xor(sum, 1, 4);
    sum += __shfl_xor(sum, 2, 4);
    if ((lane & 3) == 0) {
        const int h = lane >> 2;
        alpha[(size_t)e * 8 + h] = sum;
        atomicMax(amax + (size_t)d * 8 + h, sum);
    }
}

__global__ __launch_bounds__(256)
void edge_exp1(float* __restrict__ alpha, const float* __restrict__ amax,
               float* __restrict__ denom, const int* __restrict__ ei, int E, int N) {
    long idx = (long)blockIdx.x * blockDim.x + threadIdx.x;
    if (idx >= (long)(E + N) * 8) return;
    const int e = (int)(idx >> 3), h = (int)(idx & 7);
    const int d = (e < E) ? ei[E + e] : e - E;
    const float ea = expf(alpha[idx] - amax[(size_t)d * 8 + h]);
    alpha[idx] = ea;                       // overwrite with exp(alpha - amax)
    atomicAdd(denom + (size_t)d * 8 + h, ea);
}

__global__ __launch_bounds__(256)
void edge_aggr1(const float* __restrict__ xl, const float* __restrict__ alpha,
                const float* __restrict__ denom, const int* __restrict__ ei,
                float* __restrict__ hout, int E, int N) {
    const int e = blockIdx.x * 8 + (threadIdx.x >> 5);
    const int lane = threadIdx.x & 31;
    if (e >= E + N) return;
    int s, d;
    if (e < E) { s = ei[e]; d = ei[E + e]; } else { s = d = e - E; }
    const int h = lane >> 2;
    const float w = alpha[(size_t)e * 8 + h] / denom[(size_t)d * 8 + h];
    const int c0 = lane * 4;
    const float4 a = *(const float4*)(xl + (size_t)s * 128 + c0);
    float* pd = hout + (size_t)d * 128 + c0;
    atomicAdd(pd + 0, a.x * w);
    atomicAdd(pd + 1, a.y * w);
    atomicAdd(pd + 2, a.z * w);
    atomicAdd(pd + 3, a.w * w);
}

__global__ __launch_bounds__(256)
void bias_elu(float* __restrict__ h, const float* __restrict__ bias, long total) {
    long i = (long)blockIdx.x * blockDim.x + threadIdx.x;
    if (i >= total) return;
    const float v = h[i] + bias[i & 127];
    h[i] = v > 0.f ? v : (expf(v) - 1.f);
}

// ------------------------------------------------------- layer-2 edge passes
// 1 head, 64 channels; wave per edge, lane handles 2 channels
__global__ __launch_bounds__(256)
void edge_alpha2(const float* __restrict__ xl, const float* __restrict__ xr,
                 const int* __restrict__ ei, const float* __restrict__ att,
                 float* __restrict__ alpha, float* __restrict__ amax,
                 int E, int N) {
    const int e = blockIdx.x * 8 + (threadIdx.x >> 5);
    const int lane = threadIdx.x & 31;
    if (e >= E + N) return;
    int s, d;
    if (e < E) { s = ei[e]; d = ei[E + e]; } else { s = d = e - E; }
    const int c0 = lane * 2;
    const float2 a = *(const float2*)(xl + (size_t)s * 64 + c0);
    const float2 b = *(const float2*)(xr + (size_t)d * 64 + c0);
    const float2 t = *(const float2*)(att + c0);
    float m0 = a.x + b.x; m0 = m0 > 0.f ? m0 : NEG_SLOPE * m0;
    float m1 = a.y + b.y; m1 = m1 > 0.f ? m1 : NEG_SLOPE * m1;
    float sum = m0 * t.x + m1 * t.y;
    #pragma unroll
    for (int off = 1; off < 32; off <<= 1) sum += __shfl_xor(sum, off, 32);
    if (lane == 0) {
        alpha[e] = sum;
        atomicMax(amax + d, sum);
    }
}

__global__ __launch_bounds__(256)
void edge_exp2(float* __restrict__ alpha, const float* __restrict__ amax,
               float* __restrict__ denom, const int* __restrict__ ei, int E, int N) {
    long e = (long)blockIdx.x * blockDim.x + threadIdx.x;
    if (e >= (long)E + N) return;
    const int d = (e < E) ? ei[E + e] : (int)(e - E);
    const float ea = expf(alpha[e] - amax[d]);
    alpha[e] = ea;
    atomicAdd(denom + d, ea);
}

__global__ __launch_bounds__(256)
void edge_aggr2(const float* __restrict__ xl, const float* __restrict__ alpha,
                const float* __restrict__ denom, const int* __restrict__ ei,
                float* __restrict__ out, int E, int N) {
    const int e = blockIdx.x * 8 + (threadIdx.x >> 5);
    const int lane = threadIdx.x & 31;
    if (e >= E + N) return;
    int s, d;
    if (e < E) { s = ei[e]; d = ei[E + e]; } else { s = d = e - E; }
    const float w = alpha[e] / denom[d];
    const int c0 = lane * 2;
    const float2 a = *(const float2*)(xl + (size_t)s * 64 + c0);
    float* pd = out + (size_t)d * 64 + c0;
    atomicAdd(pd + 0, a.x * w);
    atomicAdd(pd + 1, a.y * w);
}

// --------------------------------------------------------------------- host
extern "C" void kernel_launch(void* const* d_in, const int* in_sizes, int n_in,
                              void* d_out, int out_size, void* d_ws, size_t ws_size,
                              hipStream_t stream) {
    const float* x     = (const float*)d_in[0];
    const int*   ei    = (const int*)  d_in[1];   // [2, E] int32
    const float* W1l   = (const float*)d_in[2];
    const float* b1l   = (const float*)d_in[3];
    const float* W1r   = (const float*)d_in[4];
    const float* b1r   = (const float*)d_in[5];
    const float* att1  = (const float*)d_in[6];   // [8,16] flat = channel layout
    const float* bias1 = (const float*)d_in[7];
    const float* W2l   = (const float*)d_in[8];
    const float* b2l   = (const float*)d_in[9];
    const float* W2r   = (const float*)d_in[10];
    const float* b2r   = (const float*)d_in[11];
    const float* att2  = (const float*)d_in[12];  // [1,64]
    const float* bias2 = (const float*)d_in[13];
    float* out = (float*)d_out;

    const int N    = in_sizes[0] / 128;
    const int E    = in_sizes[1] / 2;
    const int Etot = E + N;
    const int nrt  = N / 16;                       // N = 50000 -> 3125 tiles

    // workspace carve-out (floats)
    float* ws   = (float*)d_ws;
    float* xl1  = ws;  ws += (size_t)N * 128;
    float* xr1  = ws;  ws += (size_t)N * 128;
    float* hbuf = ws;  ws += (size_t)N * 128;
    float* alpha= ws;  ws += (size_t)Etot * 8;
    float* amax = ws;  ws += (size_t)N * 8;
    float* denom= ws;  ws += (size_t)N * 8;
    float* xl2  = xr1;                 // xr1 dead after layer-1 alpha pass
    float* xr2  = xr1 + (size_t)N * 64;

    const int T = 256;
    const int EB = (Etot + 7) / 8;     // wave-per-edge blocks
    const float FMIN = -3.402823466e38f;

    // ---- layer 1
    fill_f32<<<((long)N * 128 + T - 1) / T, T, 0, stream>>>(hbuf, 0.f, (long)N * 128);
    fill_f32<<<((long)N * 8 + T - 1) / T, T, 0, stream>>>(amax, FMIN, (long)N * 8);
    fill_f32<<<((long)N * 8 + T - 1) / T, T, 0, stream>>>(denom, 0.f, (long)N * 8);

    gemm_xw_bias<128><<<nrt, T, 0, stream>>>(x, W1l, b1l, xl1, nrt);
    gemm_xw_bias<128><<<nrt, T, 0, stream>>>(x, W1r, b1r, xr1, nrt);

    edge_alpha1<<<EB, T, 0, stream>>>(xl1, xr1, ei, att1, alpha, amax, E, N);
    edge_exp1<<<((long)Etot * 8 + T - 1) / T, T, 0, stream>>>(alpha, amax, denom, ei, E, N);
    edge_aggr1<<<EB, T, 0, stream>>>(xl1, alpha, denom, ei, hbuf, E, N);
    bias_elu<<<((long)N * 128 + T - 1) / T, T, 0, stream>>>(hbuf, bias1, (long)N * 128);

    // ---- layer 2
    fill_f32<<<((long)N + T - 1) / T, T, 0, stream>>>(amax, FMIN, (long)N);
    fill_f32<<<((long)N + T - 1) / T, T, 0, stream>>>(denom, 0.f, (long)N);
    init_out_bias<<<((long)N * 64 + T - 1) / T, T, 0, stream>>>(out, bias2, (long)N * 64, 64);

    gemm_xw_bias<64><<<(nrt + 1) / 2, T, 0, stream>>>(hbuf, W2l, b2l, xl2, nrt);
    gemm_xw_bias<64><<<(nrt + 1) / 2, T, 0, stream>>>(hbuf, W2r, b2r, xr2, nrt);

    edge_alpha2<<<EB, T, 0, stream>>>(xl2, xr2, ei, att2, alpha, amax, E, N);
    edge_exp2<<<((long)Etot + T - 1) / T, T, 0, stream>>>(alpha, amax, denom, ei, E, N);
    edge_aggr2<<<EB, T, 0, stream>>>(xl2, alpha, denom, ei, out, E, N);
}